// Attention_63127429317226
// MI455X (gfx1250) — compile-verified
//
#include <hip/hip_runtime.h>
#include <math.h>

#define D_DIM 1024
#define T_DIM 32768
#define NBLK 512
#define NWAVE 4
#define NTHR (NWAVE * 32)
#define RPW (T_DIM / (NBLK * NWAVE)) // 16 rows per wave

typedef __attribute__((ext_vector_type(2))) float v2f;
typedef __attribute__((ext_vector_type(8))) float v8f;

// Low 32 bits of a flat shared-aperture pointer are the LDS byte address
// (ISA 10.2).
__device__ __forceinline__ unsigned lds_lo32(const void* p) {
  return (unsigned)(unsigned long long)p;
}

// Async-copy one 4KB row (1024 f32) global->LDS: each lane issues 8 b128
// transfers at +512B strides. INST_OFFSET is added to BOTH the LDS and the
// global address (ISA 10.7), and our per-lane layouts use identical strides,
// so one LDS base VGPR + one 64-bit global address pair suffice.
__device__ __forceinline__ void async_load_row(unsigned ldsb, const float* g) {
  asm volatile(
      "global_load_async_to_lds_b128 %0, %1, off offset:0\n\t"
      "global_load_async_to_lds_b128 %0, %1, off offset:512\n\t"
      "global_load_async_to_lds_b128 %0, %1, off offset:1024\n\t"
      "global_load_async_to_lds_b128 %0, %1, off offset:1536\n\t"
      "global_load_async_to_lds_b128 %0, %1, off offset:2048\n\t"
      "global_load_async_to_lds_b128 %0, %1, off offset:2560\n\t"
      "global_load_async_to_lds_b128 %0, %1, off offset:3072\n\t"
      "global_load_async_to_lds_b128 %0, %1, off offset:3584\n\t"
      :
      : "v"(ldsb), "v"((unsigned long long)(const void*)g)
      : "memory");
}

__global__ __launch_bounds__(NTHR) void softmax_stream_kernel(
    const float* __restrict__ r_star, const float* __restrict__ q_t,
    const float* __restrict__ W, const float* __restrict__ b,
    float* __restrict__ partial) {
  // [0,1024): cw = W*r_star ; then per-wave double buffers (2KB-float each)
  __shared__ __align__(16) float smem[D_DIM + NWAVE * 2 * D_DIM];

  for (int i = threadIdx.x; i < D_DIM / 4; i += NTHR) {
    const float4 wv = ((const float4*)W)[i];
    const float4 rv = ((const float4*)r_star)[i];
    float4 c;
    c.x = wv.x * rv.x; c.y = wv.y * rv.y;
    c.z = wv.z * rv.z; c.w = wv.w * rv.w;
    ((float4*)smem)[i] = c;
  }
  __syncthreads();

  const int w = threadIdx.x >> 5;
  const int lane = threadIdx.x & 31;
  const int gw = blockIdx.x * NWAVE + w;
  const int r0 = gw * RPW;
  const float bval = b[0];

  float* buf0 = &smem[D_DIM + w * 2 * D_DIM];
  float* buf1 = buf0 + D_DIM;
  const unsigned lb0 = lds_lo32(buf0) + (unsigned)lane * 16u;
  const unsigned lb1 = lds_lo32(buf1) + (unsigned)lane * 16u;

  float4 acc[8];
#pragma unroll
  for (int j = 0; j < 8; ++j) acc[j] = make_float4(0.f, 0.f, 0.f, 0.f);

  // prologue prefetch of row 0 into buf0
  async_load_row(lb0, q_t + (size_t)r0 * D_DIM + lane * 4);

  for (int i = 0; i < RPW; ++i) {
    if (i + 1 < RPW) {
      // prefetch next row into the other buffer, then wait for the 8
      // transfers of the CURRENT row (<=8 outstanding remain = next row's).
      async_load_row((i & 1) ? lb0 : lb1,
                     q_t + (size_t)(r0 + i + 1) * D_DIM + lane * 4);
      asm volatile("s_wait_asynccnt 0x8" ::: "memory");
    } else {
      asm volatile("s_wait_asynccnt 0x0" ::: "memory");
    }

    const float4* bp = (const float4*)((i & 1) ? buf1 : buf0);
    const float4* cw = (const float4*)smem;

    float4 qv[8], ev[8];
    float lmax = -3.402823466e38f;
#pragma unroll
    for (int j = 0; j < 8; ++j) {
      const float4 qq = bp[lane + j * 32];
      const float4 cc = cw[lane + j * 32];
      float4 e;
      e.x = fmaf(cc.x, qq.x, bval);
      e.y = fmaf(cc.y, qq.y, bval);
      e.z = fmaf(cc.z, qq.z, bval);
      e.w = fmaf(cc.w, qq.w, bval);
      lmax = fmaxf(lmax, fmaxf(fmaxf(e.x, e.y), fmaxf(e.z, e.w)));
      qv[j] = qq;
      ev[j] = e;
    }
#pragma unroll
    for (int m = 16; m >= 1; m >>= 1)
      lmax = fmaxf(lmax, __shfl_xor(lmax, m, 32));

    float lsum = 0.f;
#pragma unroll
    for (int j = 0; j < 8; ++j) {
      ev[j].x = __expf(ev[j].x - lmax);
      ev[j].y = __expf(ev[j].y - lmax);
      ev[j].z = __expf(ev[j].z - lmax);
      ev[j].w = __expf(ev[j].w - lmax);
      lsum += ev[j].x + ev[j].y + ev[j].z + ev[j].w;
    }
#pragma unroll
    for (int m = 16; m >= 1; m >>= 1)
      lsum += __shfl_xor(lsum, m, 32);
    const float inv = 1.0f / lsum;

#pragma unroll
    for (int j = 0; j < 8; ++j) {
      acc[j].x = fmaf(ev[j].x * inv, qv[j].x, acc[j].x);
      acc[j].y = fmaf(ev[j].y * inv, qv[j].y, acc[j].y);
      acc[j].z = fmaf(ev[j].z * inv, qv[j].z, acc[j].z);
      acc[j].w = fmaf(ev[j].w * inv, qv[j].w, acc[j].w);
    }
  }

  // stage per-wave accumulators into this wave's (now idle) buf0 region
  {
    float4* st = (float4*)buf0;
#pragma unroll
    for (int j = 0; j < 8; ++j) st[lane + j * 32] = acc[j];
  }
  __syncthreads();

  // deterministic cross-wave combine (fixed order) -> per-block partial
  float4* pout = (float4*)(partial + (size_t)blockIdx.x * D_DIM);
  for (int i = threadIdx.x; i < D_DIM / 4; i += NTHR) {
    float4 s = make_float4(0.f, 0.f, 0.f, 0.f);
#pragma unroll
    for (int ww = 0; ww < NWAVE; ++ww) {
      const float4 v = ((const float4*)&smem[D_DIM + ww * 2 * D_DIM])[i];
      s.x += v.x; s.y += v.y; s.z += v.z; s.w += v.w;
    }
    pout[i] = s;
  }
}

// Final cross-block reduction: out[d] = sum_p partial[p][d] done as a
// ones-vector GEMV with V_WMMA_F32_16X16X4_F32. A = 16x4 all-ones (K-order
// in the A/B register layout is irrelevant since every A element is 1.0),
// B tile = partial[p..p+3][dbase..dbase+15]. D rows are all identical =
// column sums; lane n (0..15) of C-VGPR0 holds out[dbase+n].
__global__ __launch_bounds__(64) void wmma_reduce_kernel(
    const float* __restrict__ partial, float* __restrict__ out) {
  const int lane = threadIdx.x & 31;
  const int wid = blockIdx.x * 2 + (threadIdx.x >> 5);
  const int dbase = wid * 16;
  const int n = lane & 15;
  const int koff = (lane >> 4) << 1; // lanes 0-15: K=0/1 ; lanes 16-31: K=2/3

  v8f c = {0.f, 0.f, 0.f, 0.f, 0.f, 0.f, 0.f, 0.f};
  v2f a = {1.f, 1.f};
  const float* base = partial + dbase + n;

  for (int p = 0; p < NBLK; p += 4) {
    v2f bm;
    bm.x = base[(size_t)(p + koff) * D_DIM];
    bm.y = base[(size_t)(p + koff + 1) * D_DIM];
    c = __builtin_amdgcn_wmma_f32_16x16x4_f32(
        /*neg_a=*/false, a, /*neg_b=*/false, bm,
        /*c_mod=*/(short)0, c, /*reuse_a=*/false, /*reuse_b=*/false);
  }
  if (lane < 16) out[dbase + lane] = c[0];
}

extern "C" void kernel_launch(void* const* d_in, const int* in_sizes, int n_in,
                              void* d_out, int out_size, void* d_ws,
                              size_t ws_size, hipStream_t stream) {
  const float* r_star = (const float*)d_in[0];
  const float* q_t    = (const float*)d_in[1];
  const float* W      = (const float*)d_in[2];
  const float* b      = (const float*)d_in[3];
  float* out = (float*)d_out;
  float* partial = (float*)d_ws; // NBLK * D_DIM * 4 = 2 MiB

  softmax_stream_kernel<<<NBLK, NTHR, 0, stream>>>(r_star, q_t, W, b, partial);
  wmma_reduce_kernel<<<D_DIM / 16 / 2, 64, 0, stream>>>(partial, out);
}